// NormalConsistencyLoss_39256001086050
// MI455X (gfx1250) — compile-verified
//
#include <hip/hip_runtime.h>
#include <math.h>

// ---------------------------------------------------------------------------
// NormalConsistencyLoss, fused for MI455X (gfx1250, wave32, WMMA).
//
// Squared distances are produced ENTIRELY inside V_WMMA_F32_16X16X4_F32 via
// homogeneous coordinates:
//     A row m  = (-2x_i, -2y_i, -2z_i, 1)
//     B col n  = ( x_j,   y_j,   z_j,  |x_j|^2 )
//     C[m][n]  = |x_i|^2   (broadcast per row through the accumulator input)
//  => D[m][n]  = |x_i|^2 + |x_j|^2 - 2<x_i,x_j> = d2(i,j)
// so the inner loop is: load 16 candidate points -> 2 WMMAs -> stage to LDS
// -> per-lane register top-16 update. No scalar distance math at all.
// ---------------------------------------------------------------------------

typedef __attribute__((ext_vector_type(2))) float v2f;
typedef __attribute__((ext_vector_type(8))) float v8f;

#define KNN 16
#define NPTS 8192
#define NBATCH 4

__device__ __forceinline__ float3 load3(const float* __restrict__ P, int i) {
    return make_float3(P[3 * i + 0], P[3 * i + 1], P[3 * i + 2]);
}

// One wave (32 threads) per block; each wave owns 32 query rows of one batch.
__global__ void __launch_bounds__(32)
knn_normals_kernel(const float* __restrict__ pts, float* __restrict__ normals, int Nn) {
    __shared__ float tile[32 * 16];  // 2 KB: staged d2 tiles (32 rows x 16 cols)

    const int lane = threadIdx.x;          // 0..31
    const int n16  = lane & 15;            // column / row-in-tile id
    const int hi   = lane >> 4;            // half-wave select (K-pair select)
    const int b    = blockIdx.y;
    const int rowBase = blockIdx.x * 32;   // first query row of this wave

    const float* __restrict__ P = pts + (size_t)b * Nn * 3;

    // ---- A tiles: row m = (-2x, -2y, -2z, 1) ----
    float3 q0 = load3(P, rowBase + n16);
    float3 q1 = load3(P, rowBase + 16 + n16);
    v2f a0, a1;
    a0.x = hi ? -2.0f * q0.z : -2.0f * q0.x;
    a0.y = hi ? 1.0f         : -2.0f * q0.y;
    a1.x = hi ? -2.0f * q1.z : -2.0f * q1.x;
    a1.y = hi ? 1.0f         : -2.0f * q1.y;

    // |q|^2 of the query row THIS lane owns for top-k (row = rowBase + lane)
    float3 qm = load3(P, rowBase + lane);
    float sq_mine = qm.x * qm.x + qm.y * qm.y + qm.z * qm.z;

    // Accumulator inputs: C vgpr v on this lane is row m = v + 8*hi, so
    // broadcast |x_i|^2 into that layout once.
    v8f csq0, csq1;
#pragma unroll
    for (int v = 0; v < 8; ++v) {
        csq0[v] = __shfl(sq_mine, v + 8 * hi, 32);
        csq1[v] = __shfl(sq_mine, 16 + v + 8 * hi, 32);
    }

    // ---- per-lane running top-16 (ascending; slot 15 = current worst) ----
    float bd[KNN];
    int   bi[KNN];
#pragma unroll
    for (int t = 0; t < KNN; ++t) { bd[t] = 3.0e38f; bi[t] = 0; }

    // ---- sweep all candidate tiles ----
    for (int j0 = 0; j0 < Nn; j0 += 16) {
        float3 cj = load3(P, j0 + n16);                 // candidate, col n16
        float sqJ = cj.x * cj.x + cj.y * cj.y + cj.z * cj.z;
        v2f bm;
        bm.x = hi ? cj.z : cj.x;                        // (K0,K1)/(K2,K3)
        bm.y = hi ? sqJ  : cj.y;

        // D = A x B + C  ==  d2 tile, straight out of the matrix pipe
        v8f d0 = __builtin_amdgcn_wmma_f32_16x16x4_f32(false, a0, false, bm,
                                                       (short)0, csq0, false, false);
        v8f d1 = __builtin_amdgcn_wmma_f32_16x16x4_f32(false, a1, false, bm,
                                                       (short)0, csq1, false, false);

        // stage row-major into LDS (transpose C layout -> row-per-lane)
#pragma unroll
        for (int v = 0; v < 8; ++v) {
            tile[(v + 8 * hi) * 16 + n16]      = d0[v];
            tile[(16 + v + 8 * hi) * 16 + n16] = d1[v];
        }
        __syncthreads();

        // lane l consumes query row l's 16 candidates (4x ds_load_b128)
        const float4* __restrict__ rowp = (const float4*)(tile + lane * 16);
#pragma unroll
        for (int t4 = 0; t4 < 4; ++t4) {
            float4 dv = rowp[t4];
            float dvals[4] = {dv.x, dv.y, dv.z, dv.w};
#pragma unroll
            for (int u = 0; u < 4; ++u) {
                float d = dvals[u];
                if (d < bd[KNN - 1]) {
                    bd[KNN - 1] = d;
                    bi[KNN - 1] = j0 + t4 * 4 + u;
                    // single bubble pass restores sortedness
#pragma unroll
                    for (int s = KNN - 1; s > 0; --s) {
                        if (bd[s] < bd[s - 1]) {
                            float td = bd[s]; bd[s] = bd[s - 1]; bd[s - 1] = td;
                            int   ti = bi[s]; bi[s] = bi[s - 1]; bi[s - 1] = ti;
                        }
                    }
                }
            }
        }
        __syncthreads();
    }

    // ---- covariance of the 16 neighbors (single pass: E[pp^T] - mu mu^T) ----
    float sx = 0.f, sy = 0.f, sz = 0.f;
    float sxx = 0.f, sxy = 0.f, sxz = 0.f, syy = 0.f, syz = 0.f, szz = 0.f;
#pragma unroll
    for (int t = 0; t < KNN; ++t) {
        float3 p = load3(P, bi[t]);
        sx += p.x; sy += p.y; sz += p.z;
        sxx += p.x * p.x; sxy += p.x * p.y; sxz += p.x * p.z;
        syy += p.y * p.y; syz += p.y * p.z; szz += p.z * p.z;
    }
    const float ik = 1.0f / (float)KNN;
    float mx = sx * ik, my = sy * ik, mz = sz * ik;
    float a00 = sxx * ik - mx * mx;
    float a11 = syy * ik - my * my;
    float a22 = szz * ik - mz * mz;
    float a01 = sxy * ik - mx * my;
    float a02 = sxz * ik - mx * mz;
    float a12 = syz * ik - my * mz;

    // ---- smallest eigenvalue of symmetric 3x3 (trigonometric method) ----
    float p1 = a01 * a01 + a02 * a02 + a12 * a12;
    float q  = (a00 + a11 + a22) * (1.0f / 3.0f);
    float p2 = (a00 - q) * (a00 - q) + (a11 - q) * (a11 - q) +
               (a22 - q) * (a22 - q) + 2.0f * p1;
    float p  = sqrtf(p2 * (1.0f / 6.0f) + 1e-30f);
    float ipn = 1.0f / p;
    float b00 = (a00 - q) * ipn, b11 = (a11 - q) * ipn, b22 = (a22 - q) * ipn;
    float b01 = a01 * ipn, b02 = a02 * ipn, b12 = a12 * ipn;
    float detB = b00 * (b11 * b22 - b12 * b12)
               - b01 * (b01 * b22 - b12 * b02)
               + b02 * (b01 * b12 - b11 * b02);
    float r = 0.5f * detB;
    r = fminf(1.0f, fmaxf(-1.0f, r));
    float phi = acosf(r) * (1.0f / 3.0f);
    float lam = q + 2.0f * p * cosf(phi + 2.0943951023931953f);  // smallest

    // ---- eigenvector: largest cross product of rows of (A - lam I) ----
    float r0x = a00 - lam, r0y = a01,       r0z = a02;
    float r1x = a01,       r1y = a11 - lam, r1z = a12;
    float r2x = a02,       r2y = a12,       r2z = a22 - lam;

    float c01x = r0y * r1z - r0z * r1y, c01y = r0z * r1x - r0x * r1z, c01z = r0x * r1y - r0y * r1x;
    float c02x = r0y * r2z - r0z * r2y, c02y = r0z * r2x - r0x * r2z, c02z = r0x * r2y - r0y * r2x;
    float c12x = r1y * r2z - r1z * r2y, c12y = r1z * r2x - r1x * r2z, c12z = r1x * r2y - r1y * r2x;
    float n01 = c01x * c01x + c01y * c01y + c01z * c01z;
    float n02 = c02x * c02x + c02y * c02y + c02z * c02z;
    float n12 = c12x * c12x + c12y * c12y + c12z * c12z;

    float vx = c01x, vy = c01y, vz = c01z, nn = n01;
    if (n02 > nn) { vx = c02x; vy = c02y; vz = c02z; nn = n02; }
    if (n12 > nn) { vx = c12x; vy = c12y; vz = c12z; nn = n12; }
    float inv = rsqrtf(nn + 1e-30f);
    vx *= inv; vy *= inv; vz *= inv;

    const int qr = rowBase + lane;
    float* __restrict__ outn = normals + ((size_t)b * Nn + qr) * 3;
    outn[0] = vx; outn[1] = vy; outn[2] = vz;
}

__global__ void loss_init_kernel(float* __restrict__ out) {
    out[0] = 1.0f;
}

__global__ void __launch_bounds__(256)
loss_reduce_kernel(const float* __restrict__ np, const float* __restrict__ ng,
                   float* __restrict__ out, int total) {
    int i = blockIdx.x * blockDim.x + threadIdx.x;
    float d = 0.0f;
    if (i < total) {
        const float* a = np + (size_t)i * 3;
        const float* b = ng + (size_t)i * 3;
        d = a[0] * b[0] + a[1] * b[1] + a[2] * b[2];
    }
#pragma unroll
    for (int off = 16; off > 0; off >>= 1) d += __shfl_down(d, off, 32);

    __shared__ float wsum[8];
    int lane = threadIdx.x & 31;
    int wid  = threadIdx.x >> 5;
    if (lane == 0) wsum[wid] = d;
    __syncthreads();
    if (wid == 0) {
        d = (lane < 8) ? wsum[lane] : 0.0f;
#pragma unroll
        for (int off = 4; off > 0; off >>= 1) d += __shfl_down(d, off, 32);
        if (lane == 0) atomicAdd(out, -d * (1.0f / (float)(NBATCH * NPTS)));
    }
}

extern "C" void kernel_launch(void* const* d_in, const int* in_sizes, int n_in,
                              void* d_out, int out_size, void* d_ws, size_t ws_size,
                              hipStream_t stream) {
    (void)in_sizes; (void)n_in; (void)out_size; (void)ws_size;
    const float* pred = (const float*)d_in[0];
    const float* gt   = (const float*)d_in[1];
    float* out = (float*)d_out;

    float* n_pred = (float*)d_ws;                                // B*N*3 floats
    float* n_gt   = n_pred + (size_t)NBATCH * NPTS * 3;          // B*N*3 floats

    dim3 grid(NPTS / 32, NBATCH);
    knn_normals_kernel<<<grid, 32, 0, stream>>>(pred, n_pred, NPTS);
    knn_normals_kernel<<<grid, 32, 0, stream>>>(gt,   n_gt,   NPTS);

    loss_init_kernel<<<1, 1, 0, stream>>>(out);
    int total = NBATCH * NPTS;
    loss_reduce_kernel<<<(total + 255) / 256, 256, 0, stream>>>(n_pred, n_gt, out, total);
}